// SAGE_39814346834501
// MI455X (gfx1250) — compile-verified
//
#include <hip/hip_runtime.h>
#include <cstddef>

typedef float v2f __attribute__((ext_vector_type(2)));
typedef float v8f __attribute__((ext_vector_type(8)));

#define D_FEAT 512
#define LDS_STRIDE 516  // 516 % 64 = 4 -> 16 A-lanes map to distinct LDS banks

// ---------------------------------------------------------------------------
// Edge aggregation: neigh[dst] += h_src[src], deg[dst] += 1  (COO scatter-add)
// 2 edges per 256-thread block; each thread moves one float4 (128 chunks/row).
// ---------------------------------------------------------------------------
__global__ __launch_bounds__(256) void sage_aggregate(
    const float* __restrict__ hsrc, const int* __restrict__ src,
    const int* __restrict__ dst, float* __restrict__ neigh,
    float* __restrict__ deg, int E)
{
    int e = blockIdx.x * 2 + (threadIdx.x >> 7);
    if (e >= E) return;
    int lane = threadIdx.x & 127;
    int s = src[e];
    int d = dst[e];
    const float4 v = *reinterpret_cast<const float4*>(
        hsrc + (size_t)s * D_FEAT + (size_t)lane * 4);
    float* p = neigh + (size_t)d * D_FEAT + (size_t)lane * 4;
    unsafeAtomicAdd(p + 0, v.x);
    unsafeAtomicAdd(p + 1, v.y);
    unsafeAtomicAdd(p + 2, v.z);
    unsafeAtomicAdd(p + 3, v.w);
    if (lane == 0) unsafeAtomicAdd(deg + d, 1.0f);
}

// ---------------------------------------------------------------------------
// Combine: out[r] = (neigh[r] + hdst[r]) / (deg[r] + 1)
// ---------------------------------------------------------------------------
__global__ __launch_bounds__(256) void sage_combine(
    const float* __restrict__ neigh, const float* __restrict__ hdst,
    const float* __restrict__ deg, float* __restrict__ out, int n)
{
    size_t i = (size_t)blockIdx.x * blockDim.x + threadIdx.x;
    if (i >= (size_t)n * (D_FEAT / 4)) return;
    size_t row = i >> 7;
    int c = (int)(i & 127) * 4;
    float inv = 1.0f / (deg[row] + 1.0f);
    const float4 a = *reinterpret_cast<const float4*>(neigh + row * D_FEAT + c);
    const float4 b = *reinterpret_cast<const float4*>(hdst + row * D_FEAT + c);
    float4 r;
    r.x = (a.x + b.x) * inv;
    r.y = (a.y + b.y) * inv;
    r.z = (a.z + b.z) * inv;
    r.w = (a.w + b.w) * inv;
    *reinterpret_cast<float4*>(out + row * D_FEAT + c) = r;
}

// ---------------------------------------------------------------------------
// GEMM: C[M,N] = relu?(A[M,K] @ B[K,N] + bias), fp32 via V_WMMA_F32_16X16X4_F32
// One wave per 16x16 output tile; 4 waves/block share one LDS-staged A tile.
// A-frag layout (ISA 7.12.2): lane = 16*half + l holds
//   a[j] = A[rm + l][k + 2*half + j];  B mirrored with N instead of M.
// C/D layout: acc[i] = C[rm + i + 8*half][cn + l].
// ---------------------------------------------------------------------------
template <bool RELU>
__global__ __launch_bounds__(128) void sage_gemm_wmma(
    const float* __restrict__ A, const float* __restrict__ B,
    const float* __restrict__ bias, float* __restrict__ C,
    int M, int N, int K)
{
    __shared__ float As[16 * LDS_STRIDE];

    const int rm   = blockIdx.x * 16;
    const int wave = threadIdx.x >> 5;
    const int lane = threadIdx.x & 31;
    const int half = lane >> 4;
    const int l    = lane & 15;
    const int cn   = (blockIdx.y * 4 + wave) * 16;

    // Cooperatively stage the 16xK A tile (row-clamped for ragged M).
    {
        int r    = threadIdx.x >> 3;        // 0..15
        int cseg = (threadIdx.x & 7) * 64;  // 8 segments of 64 floats
        int gr   = min(rm + r, M - 1);
        const float* gsrc = A + (size_t)gr * K + cseg;
        float* ldst = As + r * LDS_STRIDE + cseg;
#pragma unroll
        for (int i = 0; i < 16; ++i) {
            *reinterpret_cast<float4*>(ldst + i * 4) =
                *reinterpret_cast<const float4*>(gsrc + i * 4);
        }
    }
    __syncthreads();

    v8f acc = {};
    const float* arow = As + l * LDS_STRIDE + 2 * half;
    const float* bcol = B + (size_t)(2 * half) * N + cn + l;
    for (int k = 0; k < K; k += 4) {
        v2f a;
        a.x = arow[k];
        a.y = arow[k + 1];
        v2f b;
        b.x = bcol[(size_t)k * N];
        b.y = bcol[(size_t)(k + 1) * N];
        acc = __builtin_amdgcn_wmma_f32_16x16x4_f32(
            false, a, false, b, (short)0, acc, false, false);
    }

    const float bv = bias[cn + l];
#pragma unroll
    for (int i = 0; i < 8; ++i) {
        int row = rm + i + 8 * half;
        float v = acc[i] + bv;
        if (RELU) v = fmaxf(v, 0.0f);
        if (row < M) C[(size_t)row * N + cn + l] = v;
    }
}

// ---------------------------------------------------------------------------
extern "C" void kernel_launch(void* const* d_in, const int* in_sizes, int n_in,
                              void* d_out, int out_size, void* d_ws, size_t ws_size,
                              hipStream_t stream)
{
    const float* feats = (const float*)d_in[0];
    const int*   src0  = (const int*)d_in[1];
    const int*   dst0  = (const int*)d_in[2];
    const int*   src1  = (const int*)d_in[3];
    const int*   dst1  = (const int*)d_in[4];
    const float* W1    = (const float*)d_in[5];
    const float* b1    = (const float*)d_in[6];
    const float* W2    = (const float*)d_in[7];
    const float* b2    = (const float*)d_in[8];
    float* out = (float*)d_out;

    const int N1 = 25000, N2 = 5000;
    const int E0 = in_sizes[1];  // 400000
    const int E1 = in_sizes[3];  // 80000
    const size_t F = D_FEAT;

    // Workspace layout (floats); regions reused across layers:
    //   neigh0 [N1*F]  -> reused as neigh1 (first N2*F)
    //   hpre0  [N1*F]  -> reused as hpre1  (first N2*F)
    //   h1     [N1*F]
    //   deg0   [N1]    -> reused as deg1 (first N2)
    float* ws     = (float*)d_ws;
    float* neigh0 = ws;
    float* hpre0  = neigh0 + (size_t)N1 * F;
    float* h1     = hpre0 + (size_t)N1 * F;
    float* deg0   = h1 + (size_t)N1 * F;

    // ---- Layer 0 ----
    hipMemsetAsync(neigh0, 0, (size_t)N1 * F * sizeof(float), stream);
    hipMemsetAsync(deg0, 0, (size_t)N1 * sizeof(float), stream);

    sage_aggregate<<<(E0 + 1) / 2, 256, 0, stream>>>(feats, src0, dst0,
                                                     neigh0, deg0, E0);

    {
        size_t work = (size_t)N1 * (F / 4);
        sage_combine<<<(unsigned)((work + 255) / 256), 256, 0, stream>>>(
            neigh0, feats, deg0, hpre0, N1);
    }

    {
        dim3 grid((N1 + 15) / 16, 512 / 64);  // (1563, 8)
        sage_gemm_wmma<true><<<grid, 128, 0, stream>>>(hpre0, W1, b1, h1,
                                                       N1, 512, 512);
    }

    // ---- Layer 1 (reuse neigh0/deg0/hpre0 regions) ----
    hipMemsetAsync(neigh0, 0, (size_t)N2 * F * sizeof(float), stream);
    hipMemsetAsync(deg0, 0, (size_t)N2 * sizeof(float), stream);

    sage_aggregate<<<(E1 + 1) / 2, 256, 0, stream>>>(h1, src1, dst1,
                                                     neigh0, deg0, E1);

    {
        size_t work = (size_t)N2 * (F / 4);
        sage_combine<<<(unsigned)((work + 255) / 256), 256, 0, stream>>>(
            neigh0, h1, deg0, hpre0, N2);
    }

    {
        dim3 grid((N2 + 15) / 16, 256 / 64);  // (313, 4)
        sage_gemm_wmma<false><<<grid, 128, 0, stream>>>(hpre0, W2, b2, out,
                                                        N2, 256, 512);
    }
}